// EdgeSet2Set_25065429139850
// MI455X (gfx1250) — compile-verified
//
#include <hip/hip_runtime.h>
#include <cmath>

#define D        128
#define BSEG     64
#define NITER    3
#define EPSF     1e-8f
#define CHUNKS   16

typedef __attribute__((ext_vector_type(2))) float v2f;
typedef __attribute__((ext_vector_type(8))) float v8f;

__device__ __forceinline__ float sigmoidf_(float x) { return 1.0f / (1.0f + expf(-x)); }

// ---------------------------------------------------------------------------
// init: zero [h0 | c | q0] (contiguous 32768 floats at ws base), fuse biases
// ---------------------------------------------------------------------------
__global__ void init_kernel(float* __restrict__ zero_region,
                            const float* __restrict__ b_ih,
                            const float* __restrict__ b_hh,
                            float* __restrict__ bias) {
    int i = blockIdx.x * blockDim.x + threadIdx.x;
    if (i < 32768) zero_region[i] = 0.0f;
    if (i < 4 * D) bias[i] = b_ih[i] + b_hh[i];
}

// ---------------------------------------------------------------------------
// segment boundaries via binary search (edge_batch is sorted)
// ---------------------------------------------------------------------------
__global__ void segstart_kernel(const int* __restrict__ seg, int E,
                                int* __restrict__ seg_start) {
    int b = threadIdx.x;
    if (b > BSEG) return;
    if (b == BSEG) { seg_start[b] = E; return; }
    int lo = 0, hi = E;
    while (lo < hi) {
        int mid = (lo + hi) >> 1;
        if (seg[mid] < b) lo = mid + 1; else hi = mid;
    }
    seg_start[b] = lo;
}

// ---------------------------------------------------------------------------
// LSTM cell via V_WMMA_F32_16X16X4_F32.
// gates[64,512] = qin[64,256] @ w_ih^T + hin[64,128] @ w_hh^T + bias
// 32 waves (4 blocks x 256 thr). Wave w owns output tile rows m0..m0+15,
// gate-dims d0..d0+15, and accumulates all 4 gate blocks (i|f|g|o) so the
// activation epilogue is fully in-register.
// A frag (16x4, MxK): lane = m + 16*(k>=2), vgpr = k&1.
// B frag (4x16, KxN): lane = n + 16*(k>=2), vgpr = k&1.
// C/D (16x16): vgpr v, lanes 0-15 -> M=v, lanes 16-31 -> M=v+8; N = lane&15.
// ---------------------------------------------------------------------------
__global__ void lstm_wmma_kernel(const float* __restrict__ qin,   // [64,256]
                                 const float* __restrict__ hin,   // [64,128]
                                 float* __restrict__ c,           // [64,128] in/out
                                 float* __restrict__ hout,        // [64,128]
                                 float* __restrict__ qout,        // [64,256]
                                 const float* __restrict__ w_ih,  // [512,256]
                                 const float* __restrict__ w_hh,  // [512,128]
                                 const float* __restrict__ bias)  // [512]
{
    const int wave = (blockIdx.x * blockDim.x + threadIdx.x) >> 5;  // 0..31
    const int lane = threadIdx.x & 31;
    const int m0 = (wave >> 3) << 4;       // 0,16,32,48
    const int d0 = (wave & 7) << 4;        // 0..112
    const int ml  = lane & 15;             // M for A-frag, N for B/C-frag
    const int khi = lane >> 4;             // K-halve select

    v8f acc_i = {}, acc_f = {}, acc_g = {}, acc_o = {};

    // ---- K over q_star (K = 256) ----
    {
        const float* arow = qin + (m0 + ml) * (2 * D);
        const float* wi0 = w_ih + (0 * D + d0 + ml) * (2 * D);
        const float* wi1 = w_ih + (1 * D + d0 + ml) * (2 * D);
        const float* wi2 = w_ih + (2 * D + d0 + ml) * (2 * D);
        const float* wi3 = w_ih + (3 * D + d0 + ml) * (2 * D);
        for (int k0 = 0; k0 < 2 * D; k0 += 4) {
            const int kk = k0 + 2 * khi;
            v2f a  = { arow[kk], arow[kk + 1] };
            v2f bi = { wi0[kk],  wi0[kk + 1] };
            v2f bf = { wi1[kk],  wi1[kk + 1] };
            v2f bg = { wi2[kk],  wi2[kk + 1] };
            v2f bo = { wi3[kk],  wi3[kk + 1] };
            acc_i = __builtin_amdgcn_wmma_f32_16x16x4_f32(false, a, false, bi, (short)0, acc_i, false, false);
            acc_f = __builtin_amdgcn_wmma_f32_16x16x4_f32(false, a, false, bf, (short)0, acc_f, false, false);
            acc_g = __builtin_amdgcn_wmma_f32_16x16x4_f32(false, a, false, bg, (short)0, acc_g, false, false);
            acc_o = __builtin_amdgcn_wmma_f32_16x16x4_f32(false, a, false, bo, (short)0, acc_o, false, false);
        }
    }
    // ---- K over h (K = 128) ----
    {
        const float* arow = hin + (m0 + ml) * D;
        const float* wh0 = w_hh + (0 * D + d0 + ml) * D;
        const float* wh1 = w_hh + (1 * D + d0 + ml) * D;
        const float* wh2 = w_hh + (2 * D + d0 + ml) * D;
        const float* wh3 = w_hh + (3 * D + d0 + ml) * D;
        for (int k0 = 0; k0 < D; k0 += 4) {
            const int kk = k0 + 2 * khi;
            v2f a  = { arow[kk], arow[kk + 1] };
            v2f bi = { wh0[kk],  wh0[kk + 1] };
            v2f bf = { wh1[kk],  wh1[kk + 1] };
            v2f bg = { wh2[kk],  wh2[kk + 1] };
            v2f bo = { wh3[kk],  wh3[kk + 1] };
            acc_i = __builtin_amdgcn_wmma_f32_16x16x4_f32(false, a, false, bi, (short)0, acc_i, false, false);
            acc_f = __builtin_amdgcn_wmma_f32_16x16x4_f32(false, a, false, bf, (short)0, acc_f, false, false);
            acc_g = __builtin_amdgcn_wmma_f32_16x16x4_f32(false, a, false, bg, (short)0, acc_g, false, false);
            acc_o = __builtin_amdgcn_wmma_f32_16x16x4_f32(false, a, false, bo, (short)0, acc_o, false, false);
        }
    }

    // ---- activation epilogue (each lane owns 8 elements) ----
    const int nn = lane & 15;
    const int mhi = lane >> 4;
    const int dd = d0 + nn;
    #pragma unroll
    for (int v = 0; v < 8; ++v) {
        const int mm = m0 + v + 8 * mhi;
        float gi = acc_i[v] + bias[0 * D + dd];
        float gf = acc_f[v] + bias[1 * D + dd];
        float gg = acc_g[v] + bias[2 * D + dd];
        float go = acc_o[v] + bias[3 * D + dd];
        float i_ = sigmoidf_(gi);
        float f_ = sigmoidf_(gf);
        float g_ = tanhf(gg);
        float o_ = sigmoidf_(go);
        float cn = f_ * c[mm * D + dd] + i_ * g_;
        c[mm * D + dd] = cn;
        float hn = o_ * tanhf(cn);
        hout[mm * D + dd] = hn;
        qout[mm * (2 * D) + dd] = hn;   // q_star[:, :D] = h
    }
}

// ---------------------------------------------------------------------------
// e[i] = dot(feat[i,:], h[seg[i],:])  — wave per edge, float4 per lane
// ---------------------------------------------------------------------------
__global__ void edge_dot_kernel(const float* __restrict__ feat,
                                const int* __restrict__ seg,
                                const float* __restrict__ h,
                                float* __restrict__ e, int E) {
    const int lane = threadIdx.x & 31;
    const int wid = (blockIdx.x * blockDim.x + threadIdx.x) >> 5;
    const int nwaves = (gridDim.x * blockDim.x) >> 5;
    for (int edge = wid; edge < E; edge += nwaves) {
        const int s = seg[edge];
        const float4 f4 = *(const float4*)(feat + (size_t)edge * D + lane * 4);
        const float4 q4 = *(const float4*)(h + s * D + lane * 4);
        float p = f4.x * q4.x + f4.y * q4.y + f4.z * q4.z + f4.w * q4.w;
        #pragma unroll
        for (int off = 16; off > 0; off >>= 1) p += __shfl_xor(p, off, 32);
        if (lane == 0) e[edge] = p;
    }
}

// ---------------------------------------------------------------------------
// per-segment softmax prep: e_max, e <- exp(e - e_max), inv_sum = 1/(sum+eps)
// block per segment; deterministic tree reductions
// ---------------------------------------------------------------------------
__global__ void softmax_seg_kernel(float* __restrict__ e,
                                   const int* __restrict__ seg_start,
                                   float* __restrict__ inv_sum) {
    __shared__ float red[256];
    const int s = blockIdx.x;
    const int st = seg_start[s], en = seg_start[s + 1];
    const int tid = threadIdx.x;

    float m = -3.402823466e38f;
    for (int i = st + tid; i < en; i += 256) m = fmaxf(m, e[i]);
    red[tid] = m; __syncthreads();
    for (int w = 128; w > 0; w >>= 1) {
        if (tid < w) red[tid] = fmaxf(red[tid], red[tid + w]);
        __syncthreads();
    }
    const float emax = red[0];
    __syncthreads();

    float sum = 0.0f;
    for (int i = st + tid; i < en; i += 256) {
        float v = expf(e[i] - emax);
        e[i] = v;
        sum += v;
    }
    red[tid] = sum; __syncthreads();
    for (int w = 128; w > 0; w >>= 1) {
        if (tid < w) red[tid] += red[tid + w];
        __syncthreads();
    }
    if (tid == 0) inv_sum[s] = 1.0f / (red[0] + EPSF);
}

// ---------------------------------------------------------------------------
// readout partials: grid = BSEG*CHUNKS blocks; thread (d, half) accumulates
// exp-weighted feat over its chunk; fixed order -> deterministic
// ---------------------------------------------------------------------------
__global__ void readout_partial_kernel(const float* __restrict__ feat,
                                       const float* __restrict__ e_exp,
                                       const int* __restrict__ seg_start,
                                       float* __restrict__ partial) {
    const int s = blockIdx.x / CHUNKS;
    const int cchunk = blockIdx.x % CHUNKS;
    const int st = seg_start[s], en = seg_start[s + 1];
    const int len = en - st;
    const int csz = (len + CHUNKS - 1) / CHUNKS;
    const int b0 = st + cchunk * csz;
    const int b1 = min(b0 + csz, en);
    const int d = threadIdx.x & (D - 1);
    const int hf = threadIdx.x >> 7;
    float acc = 0.0f;
    for (int i = b0 + hf; i < b1; i += 2)
        acc += e_exp[i] * feat[(size_t)i * D + d];
    partial[(size_t)blockIdx.x * 256 + threadIdx.x] = acc;
}

// fixed-order reduce of partials -> q_star[:, D:2D]
__global__ void readout_reduce_kernel(const float* __restrict__ partial,
                                      const float* __restrict__ inv_sum,
                                      float* __restrict__ qout) {
    const int s = blockIdx.x;
    const int d = threadIdx.x;   // 0..127
    float sum = 0.0f;
    #pragma unroll
    for (int cchunk = 0; cchunk < CHUNKS; ++cchunk)
        for (int hf = 0; hf < 2; ++hf)
            sum += partial[((size_t)(s * CHUNKS + cchunk)) * 256 + hf * D + d];
    qout[s * (2 * D) + D + d] = sum * inv_sum[s];
}

__global__ void copy_out_kernel(const float* __restrict__ src,
                                float* __restrict__ dst, int n) {
    int i = blockIdx.x * blockDim.x + threadIdx.x;
    if (i < n) dst[i] = src[i];
}

// ---------------------------------------------------------------------------
extern "C" void kernel_launch(void* const* d_in, const int* in_sizes, int n_in,
                              void* d_out, int out_size, void* d_ws, size_t ws_size,
                              hipStream_t stream) {
    const float* feat = (const float*)d_in[0];
    const int*   seg  = (const int*)d_in[1];
    const float* w_ih = (const float*)d_in[2];
    const float* w_hh = (const float*)d_in[3];
    const float* b_ih = (const float*)d_in[4];
    const float* b_hh = (const float*)d_in[5];
    const int E = in_sizes[1];

    // workspace layout (floats)
    float* ws      = (float*)d_ws;
    float* h0      = ws;                 // 64*128     } contiguous
    float* cbuf    = h0   + BSEG * D;    // 64*128     } zero
    float* q0      = cbuf + BSEG * D;    // 64*256     } region
    float* h1      = q0   + BSEG * 2 * D;
    float* q1      = h1   + BSEG * D;
    float* bias    = q1   + BSEG * 2 * D;        // 512
    float* inv_sum = bias + 4 * D;               // 64
    float* e       = inv_sum + BSEG;             // E
    float* partial = e + E;                      // BSEG*CHUNKS*256
    int*   seg_st  = (int*)(partial + BSEG * CHUNKS * 256);  // 65

    init_kernel<<<(32768 + 255) / 256, 256, 0, stream>>>(ws, b_ih, b_hh, bias);
    segstart_kernel<<<1, 128, 0, stream>>>(seg, E, seg_st);

    for (int it = 0; it < NITER; ++it) {
        float* hi_ = (it & 1) ? h1 : h0;
        float* qi_ = (it & 1) ? q1 : q0;
        float* ho_ = (it & 1) ? h0 : h1;
        float* qo_ = (it & 1) ? q0 : q1;

        lstm_wmma_kernel<<<4, 256, 0, stream>>>(qi_, hi_, cbuf, ho_, qo_,
                                                w_ih, w_hh, bias);
        edge_dot_kernel<<<4096, 256, 0, stream>>>(feat, seg, ho_, e, E);
        softmax_seg_kernel<<<BSEG, 256, 0, stream>>>(e, seg_st, inv_sum);
        readout_partial_kernel<<<BSEG * CHUNKS, 256, 0, stream>>>(feat, e, seg_st, partial);
        readout_reduce_kernel<<<BSEG, D, 0, stream>>>(partial, inv_sum, qo_);
    }

    // final q_star lives in q1 (iteration 2 output)
    copy_out_kernel<<<(out_size + 255) / 256, 256, 0, stream>>>(q1, (float*)d_out, out_size);
}